// OFTV3_45028437131815
// MI455X (gfx1250) — compile-verified
//
#include <hip/hip_runtime.h>
#include <math.h>

// ---------------- problem constants (from reference) ----------------
#define Bc    2
#define Nc    6
#define Cc    64
#define IH    112
#define IW    200
#define IHW   (IH * IW)
#define DVc   128
#define WVc   128
#define HCc   7              // int(4.0/0.5) - 1
#define CSPLIT 4             // channel quarters per spatial point
#define CPT   (Cc / CSPLIT)  // 16 channels per thread

typedef __attribute__((ext_vector_type(2))) float v2f;
typedef __attribute__((ext_vector_type(8))) float v8f;

// --------------------------------------------------------------------
// Kernel 1: calib[b,n] = ks[b,n] (3x3) @ imu2cs[b,n] (3x4)
// One wave per (b,n). Uses V_WMMA_F32_16X16X4_F32 (K=4 affine matmul).
// A (16x4 f32): lanes 0-15 hold M=0..15 with VGPR0=K0,VGPR1=K1;
//               lanes 16-31 hold K2,K3 (ISA 7.12.2 "32-bit A 16x4").
// B (4x16 f32): row k striped across 16 lanes of one VGPR (rows 0,2 in
//               v0 halves; rows 1,3 in v1 halves) — mirror of A layout.
// D (16x16 f32, 8 VGPRs): element (m,n) in lane (m>=8?16:0)+n, vgpr m%8.
// --------------------------------------------------------------------
__global__ void calib_wmma_kernel(const float* __restrict__ ks,
                                  const float* __restrict__ imu2cs,
                                  float* __restrict__ calib) {
  const int bn   = blockIdx.x;          // 0 .. B*N-1
  const int lane = threadIdx.x;         // 0 .. 31
  const float* K = ks + bn * 9;         // 3x3
  const float* E = imu2cs + bn * 12;    // 3x4

  const int m     = lane & 15;          // row (A) / col (B)
  const int khalf = (lane >> 4) << 1;   // 0 or 2

  v2f a;
  a.x = (m < 3 && khalf     < 3) ? K[m * 3 + khalf]     : 0.0f;
  a.y = (m < 3 && khalf + 1 < 3) ? K[m * 3 + khalf + 1] : 0.0f;

  v2f bb;
  bb.x = (m < 4 && khalf     < 3) ? E[khalf       * 4 + m] : 0.0f;
  bb.y = (m < 4 && khalf + 1 < 3) ? E[(khalf + 1) * 4 + m] : 0.0f;

  v8f c = {};
  c = __builtin_amdgcn_wmma_f32_16x16x4_f32(
      /*neg_a=*/false, a, /*neg_b=*/false, bb,
      /*c_mod=*/(short)0, c, /*reuse_a=*/false, /*reuse_b=*/false);

  // calib[i][j], i<3, j<4 lives in lane j, accumulator vgpr i.
  if (lane < 4) {
    calib[bn * 12 + 0 * 4 + lane] = c[0];
    calib[bn * 12 + 1 * 4 + lane] = c[1];
    calib[bn * 12 + 2 * 4 + lane] = c[2];
  }
}

// --------------------------------------------------------------------
// Kernel 2: projection + masked gather + max over cameras.
// thread = (b, h, cs, d, w), w fastest (gather locality + coalesced
// stores). 6 camera (base-pointer, visible) pairs computed once, reused
// for 16 channels.
// --------------------------------------------------------------------
__global__ void oft_gather_max_kernel(const float* __restrict__ features,
                                      const float* __restrict__ ks,
                                      const float* __restrict__ post_rots,
                                      const float* __restrict__ post_trans,
                                      const float* __restrict__ undists,
                                      const float* __restrict__ calib,
                                      float* __restrict__ out) {
  const long long total = (long long)Bc * HCc * CSPLIT * DVc * WVc;
  long long tid = (long long)blockIdx.x * blockDim.x + threadIdx.x;
  if (tid >= total) return;

  const int w  = (int)(tid % WVc);
  const int d  = (int)((tid / WVc) % DVc);
  const int cs = (int)((tid / (WVc * DVc)) % CSPLIT);
  const int h  = (int)((tid / (WVc * DVc * CSPLIT)) % HCc);
  const int b  = (int)(tid / ((long long)WVc * DVc * CSPLIT * HCc));

  // corner position: grid[b,d,w] + (0,0,zc[h])
  const float px = 0.5f * (float)d - 32.0f;
  const float py = 0.5f * (float)w - 32.0f;
  const float pz = 2.0f - 0.5f * (float)h;

  const float* fbase[Nc];
  unsigned vmask = 0u;

#pragma unroll
  for (int n = 0; n < Nc; ++n) {
    const int bn = b * Nc + n;
    const float* cal = calib + bn * 12;
    const float hx = cal[0] * px + cal[1] * py + cal[2]  * pz + cal[3];
    const float hy = cal[4] * px + cal[5] * py + cal[6]  * pz + cal[7];
    const float hz = cal[8] * px + cal[9] * py + cal[10] * pz + cal[11];
    const float front = (hz > 0.0f) ? 1.0f : 0.0f;
    const float imx = hx * front / hz;   // matches reference op order
    const float imy = hy * front / hz;

    const float* kk = ks + bn * 9;
    const float fx = kk[0], cx = kk[2], fy = kk[4], cy = kk[5];
    const float x = (imx - cx) / fx;
    const float y = (imy - cy) / fy;

    const float* u = undists + bn * 7;
    const float r2 = x * x + y * y;
    const float r  = sqrtf(r2);
    const float th = atanf(r);
    const float t2 = th * th;
    const float rad  = th * (1.0f + t2 * (u[0] + t2 * (u[1] + t2 * (u[2] + t2 * u[5])))) / r;
    const float xf = x * rad * fx + cx;
    const float yf = y * rad * fy + cy;
    const float poly = 1.0f + r2 * (u[0] + r2 * (u[1] + r2 * u[2]));
    const float xp = (x * poly + 2.0f * u[3] * x * y + u[4] * (r2 + 2.0f * x * x)) * fx + cx;
    const float yp = (y * poly + u[3] * (r2 + 2.0f * y * y) + 2.0f * u[4] * x * y) * fy + cy;
    const bool fish = (u[6] == 1.0f);
    const float xd = (fish ? xf : xp) * front;
    const float yd = (fish ? yf : yp) * front;

    const float* pr = post_rots + bn * 9;
    const float* pt = post_trans + bn * 3;
    const float ptx = pr[0] * xd + pr[1] * yd + pt[0];
    const float pty = pr[3] * xd + pr[4] * yd + pt[1];

    const float nx = fminf(fmaxf(ptx, -1073741824.0f), 1073741824.0f);
    const float ny = fminf(fmaxf(pty, -1073741824.0f), 1073741824.0f);
    const int bx = (int)nx;   // truncation toward zero, as astype(int32)
    const int by = (int)ny;
    const bool vis = (bx > 0) & (bx < IW) & (by > 0) & (by < IH);
    const int lin = min(max(by, 0), IH - 1) * IW + min(max(bx, 0), IW - 1);

    fbase[n] = features + (size_t)bn * Cc * IHW + lin;
    vmask |= (vis ? 1u : 0u) << n;
    __builtin_prefetch(fbase[n], 0, 0);   // global_prefetch_b8
  }

  const int c0 = cs * CPT;
  float* op = out + ((size_t)b * Cc * HCc + (size_t)h) * (DVc * WVc)
                  + (size_t)d * WVc + w;

#pragma unroll 4
  for (int i = 0; i < CPT; ++i) {
    const int c = c0 + i;
    const size_t off = (size_t)c * IHW;
    float mval = -INFINITY;
#pragma unroll
    for (int n = 0; n < Nc; ++n) {
      const float g = fbase[n][off];                    // unconditional load
      const float v = ((vmask >> n) & 1u) ? g : 0.0f;   // mask via select
      mval = fmaxf(mval, v);
    }
    op[(size_t)c * HCc * (DVc * WVc)] = mval;           // channel = c*HC + h
  }
}

// --------------------------------------------------------------------
extern "C" void kernel_launch(void* const* d_in, const int* in_sizes, int n_in,
                              void* d_out, int out_size, void* d_ws, size_t ws_size,
                              hipStream_t stream) {
  const float* features   = (const float*)d_in[0];
  const float* ks         = (const float*)d_in[1];
  const float* imu2cs     = (const float*)d_in[2];
  const float* post_rots  = (const float*)d_in[3];
  const float* post_trans = (const float*)d_in[4];
  const float* undists    = (const float*)d_in[5];
  // d_in[6] = grid — analytic, recomputed in-kernel.
  float* out   = (float*)d_out;
  float* calib = (float*)d_ws;   // 12 * 12 floats

  calib_wmma_kernel<<<Bc * Nc, 32, 0, stream>>>(ks, imu2cs, calib);

  const long long total = (long long)Bc * HCc * CSPLIT * DVc * WVc;  // 917504
  const int block = 256;
  const int gridsz = (int)((total + block - 1) / block);
  oft_gather_max_kernel<<<gridsz, block, 0, stream>>>(
      features, ks, post_rots, post_trans, undists, calib, out);
}